// st_grm_13554916786201
// MI455X (gfx1250) — compile-verified
//
#include <hip/hip_runtime.h>

#define Nn 32
#define Cc 128
#define Tt 256
#define Vv 16
#define TV 4096       // Tt*Vv
#define CTV 524288    // Cc*TV
#define BIGF 16777216 // Nn*CTV == Vv*Nn*Cc*Tt

typedef float v2f __attribute__((ext_vector_type(2)));
typedef float v8f __attribute__((ext_vector_type(8)));

static __device__ __forceinline__ v8f wmma4(v2f a, v2f b, v8f c) {
  // D(16x16,f32) = A(16x4,f32) x B(4x16,f32) + C   -> v_wmma_f32_16x16x4_f32
  return __builtin_amdgcn_wmma_f32_16x16x4_f32(false, a, false, b, (short)0, c,
                                               false, false);
}

// ---------------------------------------------------------------------------
// Adjacency: A_eff = softmax(relu(nv1@nv2), axis=1) * top1-mask  (16x16)
// ---------------------------------------------------------------------------
__global__ void k_adj(const float* __restrict__ nv1,
                      const float* __restrict__ nv2, float* __restrict__ A) {
  const int r = threadIdx.x;
  if (r >= 16) return;
  float p[16];
#pragma unroll
  for (int w = 0; w < 16; ++w) p[w] = 0.f;
  for (int k = 0; k < 64; ++k) {
    const float a = nv1[r * 64 + k];
#pragma unroll
    for (int w = 0; w < 16; ++w) p[w] += a * nv2[k * 16 + w];
  }
  float mx = -1e30f;
#pragma unroll
  for (int w = 0; w < 16; ++w) { p[w] = fmaxf(p[w], 0.f); mx = fmaxf(mx, p[w]); }
  float sum = 0.f;
#pragma unroll
  for (int w = 0; w < 16; ++w) { p[w] = expf(p[w] - mx); sum += p[w]; }
  const float inv = 1.f / sum;
  float m1 = -1.f, m2 = -1.f;
#pragma unroll
  for (int w = 0; w < 16; ++w) {
    p[w] *= inv;
    const float v = p[w];
    if (v > m1) { m2 = m1; m1 = v; } else if (v > m2) { m2 = v; }
  }
  const float thr = 1.f / 16.f;
#pragma unroll
  for (int w = 0; w < 16; ++w) {
    const float v = p[w];
    A[r * 16 + w] = (v > thr && v > m2) ? v : 0.f;
  }
}

// ---------------------------------------------------------------------------
// d-branch: 1x1 conv (W@x + gb) fused with node aggregation (tile @ Aadj).
// X,Y layout (n,c,t,v). Each wave: 16 output rows, K=128 weights in regs.
// ---------------------------------------------------------------------------
__global__ __launch_bounds__(256) void k_gemm_agg(
    const float* __restrict__ X, const float* __restrict__ W,
    const float* __restrict__ gb, const float* __restrict__ Aadj,
    float* __restrict__ Y) {
  __shared__ float lds[8 * 256];
  const int tid = threadIdx.x;
  const int l = tid & 31;
  const int lm = l & 15, lh = l >> 4, hi8 = lh * 8;
  const int mbase = (tid >> 5) * 16;
  const int n = blockIdx.x >> 3;
  const int tb0 = (blockIdx.x & 7) * 32;

  v2f a[32];
#pragma unroll
  for (int kk = 0; kk < 32; ++kk) {
    const float* wp = W + (mbase + lm) * Cc + kk * 4 + lh * 2;
    a[kk].x = wp[0];
    a[kk].y = wp[1];
  }
  v2f bA[4];
#pragma unroll
  for (int kk = 0; kk < 4; ++kk) {
    const int q0 = kk * 4 + lh * 2;
    bA[kk].x = Aadj[q0 * 16 + lm];
    bA[kk].y = Aadj[(q0 + 1) * 16 + lm];
  }
  float gbr[8];
#pragma unroll
  for (int r = 0; r < 8; ++r) gbr[r] = gb[mbase + r + hi8];

  const float* Xn = X + n * CTV;
  float* Yn = Y + n * CTV + mbase * TV;
  float* myLds = lds + (tid >> 5) * 256;

  for (int tt = 0; tt < 32; ++tt) {
    const int t = tb0 + tt;
    const float* xc = Xn + t * Vv + lm;
    v8f c = {0, 0, 0, 0, 0, 0, 0, 0};
#pragma unroll
    for (int kk = 0; kk < 32; ++kk) {
      const int q0 = kk * 4 + lh * 2;
      v2f b;
      b.x = xc[q0 * TV];
      b.y = xc[(q0 + 1) * TV];
      c = wmma4(a[kk], b, c);
    }
    __syncthreads();
#pragma unroll
    for (int r = 0; r < 8; ++r) myLds[(r + hi8) * 16 + lm] = c[r] + gbr[r];
    __syncthreads();
    v8f c2 = {0, 0, 0, 0, 0, 0, 0, 0};
#pragma unroll
    for (int kk = 0; kk < 4; ++kk) {
      const int q0 = kk * 4 + lh * 2;
      v2f a2;
      a2.x = myLds[lm * 16 + q0];
      a2.y = myLds[lm * 16 + q0 + 1];
      c2 = wmma4(a2, bA[kk], c2);
    }
    float* yp = Yn + t * Vv + lm;
#pragma unroll
    for (int r = 0; r < 8; ++r) yp[(r + hi8) * TV] = c2[r];
  }
}

// ---------------------------------------------------------------------------
// d-branch temporal conv kt=3 over T: Z = sum_k tw[:,:,k] @ Yin(t+k-1) + tb
// ---------------------------------------------------------------------------
__global__ __launch_bounds__(256) void k_tconv(const float* __restrict__ Yin,
                                               const float* __restrict__ tw,
                                               const float* __restrict__ tb,
                                               float* __restrict__ Z) {
  const int tid = threadIdx.x;
  const int l = tid & 31;
  const int lm = l & 15, lh = l >> 4, hi8 = lh * 8;
  const int mbase = (tid >> 5) * 16;
  const int n = blockIdx.x >> 3;
  const int tb0 = (blockIdx.x & 7) * 32;

  float tbr[8];
#pragma unroll
  for (int r = 0; r < 8; ++r) tbr[r] = tb[mbase + r + hi8];

  const float* Yn = Yin + n * CTV;
  float* Zn = Z + n * CTV + mbase * TV;
  const float* wrow = tw + (mbase + lm) * (Cc * 3);
  const v8f zero = {0, 0, 0, 0, 0, 0, 0, 0};

  for (int g = 0; g < 8; ++g) {
    v8f cg[4] = {zero, zero, zero, zero};
    for (int k = 0; k < 3; ++k) {
      v2f a[32];
#pragma unroll
      for (int kk = 0; kk < 32; ++kk) {
        const int q0 = kk * 4 + lh * 2;
        a[kk].x = wrow[q0 * 3 + k];
        a[kk].y = wrow[(q0 + 1) * 3 + k];
      }
#pragma unroll
      for (int q = 0; q < 4; ++q) {
        const int t = tb0 + g * 4 + q;
        const int tp = t + k - 1;
        if (tp >= 0 && tp < Tt) {
          const float* xc = Yn + tp * Vv + lm;
#pragma unroll
          for (int kk = 0; kk < 32; ++kk) {
            const int q0 = kk * 4 + lh * 2;
            v2f b;
            b.x = xc[q0 * TV];
            b.y = xc[(q0 + 1) * TV];
            cg[q] = wmma4(a[kk], b, cg[q]);
          }
        }
      }
    }
#pragma unroll
    for (int q = 0; q < 4; ++q) {
      float* zp = Zn + (tb0 + g * 4 + q) * Vv + lm;
#pragma unroll
      for (int r = 0; r < 8; ++r) zp[(r + hi8) * TV] = cg[q][r] + tbr[r];
    }
  }
}

// ---------------------------------------------------------------------------
// static 1x1 conv. mode 0: read x(n,c,s,p); mode 1: read (p,n,c,s).
// Out layout (p,n,o,s). One block per (p,n).
// ---------------------------------------------------------------------------
__global__ __launch_bounds__(256) void k_sgemm(const float* __restrict__ In,
                                               const float* __restrict__ W,
                                               float* __restrict__ Out,
                                               int mode) {
  const int tid = threadIdx.x;
  const int l = tid & 31;
  const int lm = l & 15, lh = l >> 4, hi8 = lh * 8;
  const int mbase = (tid >> 5) * 16;
  const int p = blockIdx.x >> 5;
  const int n = blockIdx.x & 31;

  v2f a[32];
#pragma unroll
  for (int kk = 0; kk < 32; ++kk) {
    const float* wp = W + (mbase + lm) * Cc + kk * 4 + lh * 2;
    a[kk].x = wp[0];
    a[kk].y = wp[1];
  }
  const float* xb0 = In + n * CTV + p;              // mode 0: x[n,c,s,p]
  const float* xb1 = In + ((p * Nn + n) * Cc) * Tt; // mode 1: (p,n,c,s)
  float* ob = Out + ((p * Nn + n) * Cc + mbase) * Tt;

  for (int st = 0; st < 16; ++st) {
    const int scol = st * 16 + lm;
    v8f c = {0, 0, 0, 0, 0, 0, 0, 0};
    if (mode == 0) {
#pragma unroll
      for (int kk = 0; kk < 32; ++kk) {
        const int q0 = kk * 4 + lh * 2;
        v2f b;
        b.x = xb0[q0 * TV + scol * Vv];
        b.y = xb0[(q0 + 1) * TV + scol * Vv];
        c = wmma4(a[kk], b, c);
      }
    } else {
#pragma unroll
      for (int kk = 0; kk < 32; ++kk) {
        const int q0 = kk * 4 + lh * 2;
        v2f b;
        b.x = xb1[q0 * Tt + scol];
        b.y = xb1[(q0 + 1) * Tt + scol];
        c = wmma4(a[kk], b, c);
      }
    }
    float* yp = ob + scol;
#pragma unroll
    for (int r = 0; r < 8; ++r) yp[(r + hi8) * Tt] = c[r];
  }
}

// ---------------------------------------------------------------------------
// static grouped temporal conv: per node p its own (128,128,3) weights.
// In/Out layout (p,n,c,s). Per-lane boundary predication on s.
// ---------------------------------------------------------------------------
__global__ __launch_bounds__(256) void k_stconv(const float* __restrict__ In,
                                                const float* __restrict__ tw,
                                                float* __restrict__ Z) {
  const int tid = threadIdx.x;
  const int l = tid & 31;
  const int lm = l & 15, lh = l >> 4, hi8 = lh * 8;
  const int mbase = (tid >> 5) * 16;
  const int p = blockIdx.x >> 5;
  const int n = blockIdx.x & 31;

  const float* ib = In + ((p * Nn + n) * Cc) * Tt;
  float* ob = Z + ((p * Nn + n) * Cc + mbase) * Tt;
  const float* wrow = tw + (p * Cc + mbase + lm) * (Cc * 3);
  const v8f zero = {0, 0, 0, 0, 0, 0, 0, 0};

  for (int g = 0; g < 4; ++g) {
    v8f cg[4] = {zero, zero, zero, zero};
    for (int k = 0; k < 3; ++k) {
      v2f a[32];
#pragma unroll
      for (int kk = 0; kk < 32; ++kk) {
        const int q0 = kk * 4 + lh * 2;
        a[kk].x = wrow[q0 * 3 + k];
        a[kk].y = wrow[(q0 + 1) * 3 + k];
      }
#pragma unroll
      for (int q = 0; q < 4; ++q) {
        const int sp = g * 64 + q * 16 + lm + k - 1;
        const bool ok = (sp >= 0) && (sp < Tt);
        const int spc = ok ? sp : 0;
#pragma unroll
        for (int kk = 0; kk < 32; ++kk) {
          const int q0 = kk * 4 + lh * 2;
          v2f b;
          b.x = ok ? ib[q0 * Tt + spc] : 0.f;
          b.y = ok ? ib[(q0 + 1) * Tt + spc] : 0.f;
          cg[q] = wmma4(a[kk], b, cg[q]);
        }
      }
    }
#pragma unroll
    for (int q = 0; q < 4; ++q) {
      float* zp = ob + g * 64 + q * 16 + lm;
#pragma unroll
      for (int r = 0; r < 8; ++r) zp[(r + hi8) * Tt] = cg[q][r];
    }
  }
}

// ---------------------------------------------------------------------------
// BN pass 1: per-channel partial sum/sumsq.  addr = base(c) + o*OS + i
// base(c) = (c&127)*CS + (c>>7)*PS, i in [0, 2^Ish). 2048 elems per block.
// ---------------------------------------------------------------------------
__global__ __launch_bounds__(256) void k_bnsum(const float* __restrict__ In,
                                               float* __restrict__ part, int CS,
                                               int PS, int OS, int Ish,
                                               int nchunk) {
  const int c = blockIdx.x / nchunk;
  const int ch = blockIdx.x % nchunk;
  const int base = (c & 127) * CS + (c >> 7) * PS;
  const int im = (1 << Ish) - 1;
  float s = 0.f, s2 = 0.f;
  for (int j = 0; j < 8; ++j) {
    const int flat = ch * 2048 + j * 256 + threadIdx.x;
    const float v = In[base + (flat >> Ish) * OS + (flat & im)];
    s += v;
    s2 += v * v;
  }
  __shared__ float ls[256], ls2[256];
  const int tid = threadIdx.x;
  ls[tid] = s;
  ls2[tid] = s2;
  __syncthreads();
  for (int st = 128; st > 0; st >>= 1) {
    if (tid < st) {
      ls[tid] += ls[tid + st];
      ls2[tid] += ls2[tid + st];
    }
    __syncthreads();
  }
  if (tid == 0) {
    part[(c * nchunk + ch) * 2] = ls[0];
    part[(c * nchunk + ch) * 2 + 1] = ls2[0];
  }
}

// BN pass 2: fold mean/var + gamma/beta into affine scale/shift.
__global__ void k_bnfin(const float* __restrict__ part, int nchunk,
                        float inv_count, const float* __restrict__ g,
                        const float* __restrict__ b, float* __restrict__ scale,
                        float* __restrict__ shift, int nchan) {
  const int c = blockIdx.x * blockDim.x + threadIdx.x;
  if (c >= nchan) return;
  float s = 0.f, s2 = 0.f;
  for (int j = 0; j < nchunk; ++j) {
    s += part[(c * nchunk + j) * 2];
    s2 += part[(c * nchunk + j) * 2 + 1];
  }
  const float m = s * inv_count;
  float var = s2 * inv_count - m * m;
  var = fmaxf(var, 0.f);
  const float sc = g[c] * rsqrtf(var + 1e-5f);
  scale[c] = sc;
  shift[c] = b[c] - m * sc;
}

// In-place affine (+optional relu). ch = ((i>>sh1)&127) + (((i>>sh2)&m2)<<7)
__global__ __launch_bounds__(256) void k_affine(float* __restrict__ buf,
                                                const float* __restrict__ scale,
                                                const float* __restrict__ shift,
                                                int sh1, int sh2, int m2,
                                                int relu, int ntot) {
  const int i = blockIdx.x * blockDim.x + threadIdx.x;
  if (i >= ntot) return;
  const int c = ((i >> sh1) & 127) + (((i >> sh2) & m2) << 7);
  float v = scale[c] * buf[i] + shift[c];
  if (relu) v = fmaxf(v, 0.f);
  buf[i] = v;
}

// out = relu(scale[c]*z + shift[c] + x)   (d-branch residual), c=(i>>12)&127
__global__ __launch_bounds__(256) void k_res_relu(
    const float* __restrict__ z, const float* __restrict__ x,
    const float* __restrict__ scale, const float* __restrict__ shift,
    float* __restrict__ out, int ntot) {
  const int i = blockIdx.x * blockDim.x + threadIdx.x;
  if (i >= ntot) return;
  const int c = (i >> 12) & 127;
  out[i] = fmaxf(scale[c] * z[i] + shift[c] + x[i], 0.f);
}

// d-branch head: out[n,o,v] = max_t relu(sc[o]*z2 + sh[o] + d1)
__global__ __launch_bounds__(256) void k_dmax(const float* __restrict__ z2,
                                              const float* __restrict__ d1,
                                              const float* __restrict__ scale,
                                              const float* __restrict__ shift,
                                              float* __restrict__ out) {
  const int n = blockIdx.x >> 7;
  const int o = blockIdx.x & 127;
  const int tid = threadIdx.x;
  const int v = tid & 15, tg = tid >> 4;
  const float* zp = z2 + (n * Cc + o) * TV;
  const float* dp = d1 + (n * Cc + o) * TV;
  const float sc = scale[o], sh = shift[o];
  float m = 0.f;  // relu output >= 0
  for (int j = 0; j < 16; ++j) {
    const int idx = (tg * 16 + j) * Vv + v;
    m = fmaxf(m, fmaxf(sc * zp[idx] + sh + dp[idx], 0.f));
  }
  __shared__ float ls[256];
  ls[tid] = m;
  __syncthreads();
  for (int st = 128; st >= 16; st >>= 1) {
    if (tid < st) ls[tid] = fmaxf(ls[tid], ls[tid + st]);
    __syncthreads();
  }
  if (tid < 16) out[(n * Cc + o) * 32 + tid] = ls[tid];
}

// static head: out[n,o,16+p] = max_s (sc[p*128+o]*zs2[p,n,o,s] + sh)
__global__ __launch_bounds__(256) void k_smax(const float* __restrict__ zs2,
                                              const float* __restrict__ scale,
                                              const float* __restrict__ shift,
                                              float* __restrict__ out) {
  const int n = blockIdx.x >> 7;
  const int o = blockIdx.x & 127;
  const int tid = threadIdx.x;
  __shared__ float ls[256];
  for (int p = 0; p < 16; ++p) {
    const float sc = scale[p * Cc + o], sh = shift[p * Cc + o];
    ls[tid] = sc * zs2[((p * Nn + n) * Cc + o) * Tt + tid] + sh;
    __syncthreads();
    for (int st = 128; st > 0; st >>= 1) {
      if (tid < st) ls[tid] = fmaxf(ls[tid], ls[tid + st]);
      __syncthreads();
    }
    if (tid == 0) out[(n * Cc + o) * 32 + 16 + p] = ls[0];
    __syncthreads();
  }
}

// ---------------------------------------------------------------------------
extern "C" void kernel_launch(void* const* d_in, const int* in_sizes, int n_in,
                              void* d_out, int out_size, void* d_ws,
                              size_t ws_size, hipStream_t stream) {
  const float* x = (const float*)d_in[0];
  const float* nv1 = (const float*)d_in[1];
  const float* nv2 = (const float*)d_in[2];
  const float* d1gw = (const float*)d_in[3];
  const float* d1gb = (const float*)d_in[4];
  const float* d1b1g = (const float*)d_in[5];
  const float* d1b1b = (const float*)d_in[6];
  const float* d1tw = (const float*)d_in[7];
  const float* d1tb = (const float*)d_in[8];
  const float* d1b2g = (const float*)d_in[9];
  const float* d1b2b = (const float*)d_in[10];
  const float* d2gw = (const float*)d_in[11];
  const float* d2gb = (const float*)d_in[12];
  const float* d2b1g = (const float*)d_in[13];
  const float* d2b1b = (const float*)d_in[14];
  const float* d2tw = (const float*)d_in[15];
  const float* d2tb = (const float*)d_in[16];
  const float* d2b2g = (const float*)d_in[17];
  const float* d2b2b = (const float*)d_in[18];
  const float* s1pw = (const float*)d_in[19];
  const float* s1pg = (const float*)d_in[20];
  const float* s1pb = (const float*)d_in[21];
  const float* s1tw = (const float*)d_in[22];
  const float* s1tg = (const float*)d_in[23];
  const float* s1tb = (const float*)d_in[24];
  const float* s2pw = (const float*)d_in[25];
  const float* s2pg = (const float*)d_in[26];
  const float* s2pb = (const float*)d_in[27];
  const float* s2tw = (const float*)d_in[28];
  const float* s2tg = (const float*)d_in[29];
  const float* s2tb = (const float*)d_in[30];
  float* out = (float*)d_out;
  (void)in_sizes; (void)n_in; (void)out_size;

  const size_t need = (size_t)(32768 + 3 * (size_t)BIGF) * sizeof(float);
  if (ws_size < need) return;

  float* wsf = (float*)d_ws;
  float* A = wsf;            // 256
  float* scale = wsf + 256;  // 2048
  float* shift = wsf + 2304; // 2048
  float* part = wsf + 4352;  // 16384
  float* B0 = wsf + 32768;
  float* B1 = B0 + BIGF;
  float* B2 = B1 + BIGF;

  const float invBN = 1.f / 131072.f;  // d-branch & static BN1 counts
  const float invBN2 = 1.f / 8192.f;   // static per-(p,c) BN count

  // ---- adjacency ----
  k_adj<<<1, 32, 0, stream>>>(nv1, nv2, A);

  // ---- d-branch block 1 ----
  k_gemm_agg<<<256, 256, 0, stream>>>(x, d1gw, d1gb, A, B0);
  k_bnsum<<<128 * 64, 256, 0, stream>>>(B0, part, 4096, 0, CTV, 12, 64);
  k_bnfin<<<1, 128, 0, stream>>>(part, 64, invBN, d1b1g, d1b1b, scale, shift, 128);
  k_affine<<<BIGF / 256, 256, 0, stream>>>(B0, scale, shift, 12, 20, 0, 1, BIGF);
  k_tconv<<<256, 256, 0, stream>>>(B0, d1tw, d1tb, B1);
  k_bnsum<<<128 * 64, 256, 0, stream>>>(B1, part, 4096, 0, CTV, 12, 64);
  k_bnfin<<<1, 128, 0, stream>>>(part, 64, invBN, d1b2g, d1b2b, scale, shift, 128);
  k_res_relu<<<BIGF / 256, 256, 0, stream>>>(B1, x, scale, shift, B2, BIGF);

  // ---- d-branch block 2 ----
  k_gemm_agg<<<256, 256, 0, stream>>>(B2, d2gw, d2gb, A, B0);
  k_bnsum<<<128 * 64, 256, 0, stream>>>(B0, part, 4096, 0, CTV, 12, 64);
  k_bnfin<<<1, 128, 0, stream>>>(part, 64, invBN, d2b1g, d2b1b, scale, shift, 128);
  k_affine<<<BIGF / 256, 256, 0, stream>>>(B0, scale, shift, 12, 20, 0, 1, BIGF);
  k_tconv<<<256, 256, 0, stream>>>(B0, d2tw, d2tb, B1);
  k_bnsum<<<128 * 64, 256, 0, stream>>>(B1, part, 4096, 0, CTV, 12, 64);
  k_bnfin<<<1, 128, 0, stream>>>(part, 64, invBN, d2b2g, d2b2b, scale, shift, 128);
  k_dmax<<<Nn * Cc, 256, 0, stream>>>(B1, B2, scale, shift, out);

  // ---- static block 1 ----
  k_sgemm<<<512, 256, 0, stream>>>(x, s1pw, B0, 0);
  k_bnsum<<<128 * 64, 256, 0, stream>>>(B0, part, 256, 0, 32768, 8, 64);
  k_bnfin<<<1, 128, 0, stream>>>(part, 64, invBN, s1pg, s1pb, scale, shift, 128);
  k_affine<<<BIGF / 256, 256, 0, stream>>>(B0, scale, shift, 8, 20, 0, 0, BIGF);
  k_stconv<<<512, 256, 0, stream>>>(B0, s1tw, B1);
  k_bnsum<<<2048 * 4, 256, 0, stream>>>(B1, part, 256, 1048576, 32768, 8, 4);
  k_bnfin<<<8, 256, 0, stream>>>(part, 4, invBN2, s1tg, s1tb, scale, shift, 2048);
  k_affine<<<BIGF / 256, 256, 0, stream>>>(B1, scale, shift, 8, 20, 15, 0, BIGF);

  // ---- static block 2 ----
  k_sgemm<<<512, 256, 0, stream>>>(B1, s2pw, B0, 1);
  k_bnsum<<<128 * 64, 256, 0, stream>>>(B0, part, 256, 0, 32768, 8, 64);
  k_bnfin<<<1, 128, 0, stream>>>(part, 64, invBN, s2pg, s2pb, scale, shift, 128);
  k_affine<<<BIGF / 256, 256, 0, stream>>>(B0, scale, shift, 8, 20, 0, 0, BIGF);
  k_stconv<<<512, 256, 0, stream>>>(B0, s2tw, B1);
  k_bnsum<<<2048 * 4, 256, 0, stream>>>(B1, part, 256, 1048576, 32768, 8, 4);
  k_bnfin<<<8, 256, 0, stream>>>(part, 4, invBN2, s2tg, s2tb, scale, shift, 2048);
  k_smax<<<Nn * Cc, 256, 0, stream>>>(B1, scale, shift, out);
}